// WSoftmax_52029233824325
// MI455X (gfx1250) — compile-verified
//
#include <hip/hip_runtime.h>
#include <hip/hip_bf16.h>

#define ALPHA_F 1.5f
#define EPS_F   1e-12f
#define C_CLS   1000
#define C_PAD   1008   // 63 * 16
#define D_DIM   512
#define N_SMP   256

typedef __attribute__((ext_vector_type(2))) float v2f;
typedef __attribute__((ext_vector_type(8))) float v8f;

// ---------------------------------------------------------------------------
// Kernel 1: row-wise L2 normalize W (C_CLS x D_DIM) -> Wn (C_PAD x D_DIM).
// Pad rows [1000,1008) are zeroed so the GEMM can load B unconditionally.
// ---------------------------------------------------------------------------
__global__ __launch_bounds__(256) void wnorm_kernel(const float* __restrict__ W,
                                                    float* __restrict__ Wn) {
  const int c   = blockIdx.x;
  const int tid = threadIdx.x;
  if (c >= C_CLS) {
    v2f z = {0.0f, 0.0f};
    *(v2f*)(Wn + (size_t)c * D_DIM + 2 * tid) = z;
    return;
  }
  __shared__ float red[256];
  v2f v = *(const v2f*)(W + (size_t)c * D_DIM + 2 * tid);
  red[tid] = v.x * v.x + v.y * v.y;
  __syncthreads();
  for (int s = 128; s > 0; s >>= 1) {
    if (tid < s) red[tid] += red[tid + s];
    __syncthreads();
  }
  const float inv = 1.0f / fmaxf(sqrtf(red[0]), EPS_F);
  v.x *= inv; v.y *= inv;
  *(v2f*)(Wn + (size_t)c * D_DIM + 2 * tid) = v;
}

// ---------------------------------------------------------------------------
// Kernel 2: P = A @ Wn^T with V_WMMA_F32_16X16X4_F32 (fp32 in / fp32 out).
//   A rows 0..255   = x[n]                (original logits rows)
//   A rows 256..511 = Wn[labels[n]]       (G rows: <Wc, Wn_c>)
// One wave computes one 16x16 tile; K=512 -> 128 WMMA ops per tile.
// A fragment (ISA 7.12.2, 32-bit 16x4): lane<16 -> M=lane, K=k+{0,1};
// lane>=16 -> M=lane-16, K=k+{2,3}. B mirrors with N=lane&15.
// ---------------------------------------------------------------------------
__global__ __launch_bounds__(128) void gemm_kernel(const float* __restrict__ x,
                                                   const float* __restrict__ Wn,
                                                   const int*   __restrict__ labels,
                                                   float* __restrict__ P) {
  const int lane = threadIdx.x & 31;
  const int wave = threadIdx.x >> 5;
  const int mt   = blockIdx.y * 4 + wave;   // 0..31  (row tiles, rows 0..511)
  const int ct   = blockIdx.x;              // 0..62  (class tiles)
  const int m0   = mt * 16;
  const int c0   = ct * 16;
  const int r    = lane & 15;
  const int half = lane >> 4;

  // Resolve this lane's A row once (loop-invariant); label indirection hoisted.
  const float* abase;
  int arow;
  if (m0 < N_SMP) { abase = x;  arow = m0 + r; }
  else            { abase = Wn; arow = labels[m0 - N_SMP + r]; }
  const float* aptr = abase + (size_t)arow * D_DIM + 2 * half;
  const float* bptr = Wn + (size_t)(c0 + r) * D_DIM + 2 * half;

  v8f acc = {0.f, 0.f, 0.f, 0.f, 0.f, 0.f, 0.f, 0.f};
#pragma unroll 8
  for (int k = 0; k < D_DIM; k += 4) {
    v2f a = *(const v2f*)(aptr + k);
    v2f b = *(const v2f*)(bptr + k);
    acc = __builtin_amdgcn_wmma_f32_16x16x4_f32(
        /*neg_a=*/false, a, /*neg_b=*/false, b,
        /*c_mod=*/(short)0, acc, /*reuse_a=*/false, /*reuse_b=*/false);
  }

  // C/D layout: VGPR j -> row (m0 + j + 8*half), col (c0 + r)
  float* pout = P + (size_t)(m0 + 8 * half) * C_PAD + (c0 + r);
#pragma unroll
  for (int j = 0; j < 8; ++j) pout[(size_t)j * C_PAD] = acc[j];
}

// ---------------------------------------------------------------------------
// Kernel 3: per-sample adjusted logits + log-softmax -> rowloss[n].
// adjusted[n,c] = (a*t_n + logits[n,c]) / sqrt(a^2 + 1 + 2a*G[n,c]),
// except c == label where the original logit is kept (t_n itself).
// ---------------------------------------------------------------------------
__global__ __launch_bounds__(256) void rowloss_kernel(const float* __restrict__ P,
                                                      const int*   __restrict__ labels,
                                                      float* __restrict__ rowloss) {
  const int n   = blockIdx.x;
  const int tid = threadIdx.x;
  const int lab = labels[n];
  const float tn   = P[(size_t)n * C_PAD + lab];
  const float a2p1 = ALPHA_F * ALPHA_F + 1.0f;
  __shared__ float red[256];

  float f[4];
  float m = -INFINITY;
#pragma unroll
  for (int j = 0; j < 4; ++j) {
    const int c = tid + 256 * j;
    float fj = -INFINITY;
    if (c < C_CLS) {
      const float orig = P[(size_t)n * C_PAD + c];
      if (c == lab) {
        fj = orig;
      } else {
        const float g = P[(size_t)(N_SMP + n) * C_PAD + c];
        fj = (ALPHA_F * tn + orig) / sqrtf(a2p1 + 2.0f * ALPHA_F * g);
      }
    }
    f[j] = fj;
    m = fmaxf(m, fj);
  }

  red[tid] = m;
  __syncthreads();
  for (int s = 128; s > 0; s >>= 1) {
    if (tid < s) red[tid] = fmaxf(red[tid], red[tid + s]);
    __syncthreads();
  }
  const float maxv = red[0];
  __syncthreads();

  float sum = 0.0f;
#pragma unroll
  for (int j = 0; j < 4; ++j) sum += expf(f[j] - maxv);  // exp(-inf)=0 for pads
  red[tid] = sum;
  __syncthreads();
  for (int s = 128; s > 0; s >>= 1) {
    if (tid < s) red[tid] += red[tid + s];
    __syncthreads();
  }
  if (tid == 0) rowloss[n] = -(tn - maxv - logf(red[0]));
}

// ---------------------------------------------------------------------------
// Kernel 4: mean over 256 row losses -> d_out[0]. Deterministic tree sum.
// ---------------------------------------------------------------------------
__global__ __launch_bounds__(256) void meanloss_kernel(const float* __restrict__ rowloss,
                                                       float* __restrict__ out) {
  __shared__ float red[256];
  const int tid = threadIdx.x;
  red[tid] = rowloss[tid];
  __syncthreads();
  for (int s = 128; s > 0; s >>= 1) {
    if (tid < s) red[tid] += red[tid + s];
    __syncthreads();
  }
  if (tid == 0) out[0] = red[0] * (1.0f / (float)N_SMP);
}

// ---------------------------------------------------------------------------
extern "C" void kernel_launch(void* const* d_in, const int* in_sizes, int n_in,
                              void* d_out, int out_size, void* d_ws, size_t ws_size,
                              hipStream_t stream) {
  const float* W      = (const float*)d_in[0];   // (1000, 512)
  const float* x      = (const float*)d_in[1];   // (256, 512)
  const int*   labels = (const int*)d_in[2];     // (256,)

  float* Wn      = (float*)d_ws;                        // C_PAD * D_DIM
  float* P       = Wn + (size_t)C_PAD * D_DIM;          // (2*N_SMP) * C_PAD
  float* rowloss = P + (size_t)(2 * N_SMP) * C_PAD;     // N_SMP
  float* out     = (float*)d_out;

  wnorm_kernel  <<<C_PAD, 256, 0, stream>>>(W, Wn);
  gemm_kernel   <<<dim3(63, 8), 128, 0, stream>>>(x, Wn, labels, P);
  rowloss_kernel<<<N_SMP, 256, 0, stream>>>(P, labels, rowloss);
  meanloss_kernel<<<1, 256, 0, stream>>>(rowloss, out);
}